// DecoderCrossAttention_58497454572236
// MI455X (gfx1250) — compile-verified
//
#include <hip/hip_runtime.h>
#include <math.h>

// Problem constants
#define BATCH 2
#define CONDN 4
#define CCH   128
#define NHEAD 8
#define DHEAD 16
#define VOX   (32*32*32)        // 32768 voxels per (b, channel)
#define GROUPS 8
#define GSZ   (CCH/GROUPS)      // 16 channels per group (== DHEAD == N-tile)
#define EPS   1e-5f

#define WAVES 4
#define MTILE 16
#define VOX_PER_BLOCK (WAVES*MTILE)           // 64
#define BLOCKS_PER_B  (VOX/VOX_PER_BLOCK)     // 512
#define NBLOCKS       (BATCH*BLOCKS_PER_B)    // 1024
#define LDSPITCH 132                           // 128 + 4 pad -> conflict-free

typedef __attribute__((ext_vector_type(2))) float v2f;
typedef __attribute__((ext_vector_type(8))) float v8f;

// ---------------------------------------------------------------------------
// Fragment helpers.  Layouts per ISA 7.12.2 (V_WMMA_F32_16X16X4_F32):
//   A frag: lane L -> m = L&15, K = k0 + 2*(L>>4) + {0,1}
//   B frag: lane L -> n = L&15, same K split
//   C/D   : lane L -> n col = L&15, vgpr r -> M row = r + 8*(L>>4)
// Fragment arrays are indexed only from fully-unrolled loops so they stay in
// distinct VGPRs -> loads pipeline (incremental s_wait_loadcnt, no serialize).
// ---------------------------------------------------------------------------

// One 16x16x128 tile: 32 WMMAs split into TWO independent accumulator chains
// (even/odd K-steps) so the matrix pipe sees 2-way ILP instead of one
// 32-deep serial D->C dependency chain; combined with 8 v_add_f32 at the end.
__device__ __forceinline__ v8f gemm_frags(const v2f* a, const v2f* b) {
  v8f acc0 = {};
  v8f acc1 = {};
#pragma unroll
  for (int t = 0; t < 32; t += 2) {
    acc0 = __builtin_amdgcn_wmma_f32_16x16x4_f32(false, a[t], false, b[t],
                                                 (short)0, acc0, false, false);
    acc1 = __builtin_amdgcn_wmma_f32_16x16x4_f32(false, a[t + 1], false, b[t + 1],
                                                 (short)0, acc1, false, false);
  }
  v8f acc;
#pragma unroll
  for (int r = 0; r < 8; ++r) acc[r] = acc0[r] + acc1[r];
  return acc;
}

// Activation tile A[m][k] = A[k*VOX + m] (channels-first, voxels contiguous).
__device__ __forceinline__ void load_act_frags(const float* __restrict__ A,
                                               int m, int hi, v2f* a) {
#pragma unroll
  for (int t = 0; t < 32; ++t) {
    const int c = 4 * t + 2 * hi;
    v2f v;
    v.x = A[(size_t)c * VOX + m];
    v.y = A[(size_t)(c + 1) * VOX + m];
    a[t] = v;
  }
}

// Weight tile B[k][n] = W[n][k] (row-major [C][C], computes X @ W^T).
__device__ __forceinline__ void load_w_frags(const float* __restrict__ W,
                                             int co0, int m, int hi, v2f* b) {
#pragma unroll
  for (int t = 0; t < 32; ++t) {
    const int c = 4 * t + 2 * hi;
    b[t] = *(const v2f*)(W + (co0 + m) * CCH + c);
  }
}

// A tile from LDS stash: A[m][k] = base[m*LDSPITCH + k] (8B aligned: c even).
__device__ __forceinline__ void load_lds_frags(const float* base,
                                               int m, int hi, v2f* a) {
#pragma unroll
  for (int t = 0; t < 32; ++t) {
    const int c = 4 * t + 2 * hi;
    a[t] = *(const v2f*)(base + m * LDSPITCH + c);
  }
}

// Sum within each 16-lane half of a wave32 (masks 1..8 never cross halves).
__device__ __forceinline__ float red_half16(float x) {
  x += __shfl_xor(x, 1, 32);
  x += __shfl_xor(x, 2, 32);
  x += __shfl_xor(x, 4, 32);
  x += __shfl_xor(x, 8, 32);
  return x;
}
__device__ __forceinline__ float red_wave32(float x) {
  x = red_half16(x);
  x += __shfl_xor(x, 16, 32);
  return x;
}

// ---------------------------------------------------------------------------
// Kernel 1: fused QKV projection + 4-key softmax attention + output projection
// + bias + residual.  Block = 128 threads = 4 waves; each wave owns 16 voxels.
// Each skip tile is loaded twice per wave (K, V phases), dec twice (Q frags, residual);
// q tiles / head outputs / scores flow through wave-private LDS stashes.
// ---------------------------------------------------------------------------
__global__ __launch_bounds__(128) void fused_attn_kernel(
    const float* __restrict__ skip, const float* __restrict__ dec,
    const float* __restrict__ wq, const float* __restrict__ wk,
    const float* __restrict__ wv, const float* __restrict__ bq,
    const float* __restrict__ bk, const float* __restrict__ bv,
    const float* __restrict__ wo, const float* __restrict__ bo,
    float* __restrict__ out, float* __restrict__ partials) {
  __shared__ float lds_q[WAVES][MTILE * LDSPITCH];          // q stash
  __shared__ float lds_o[WAVES][MTILE * LDSPITCH];          // attn out stash
  __shared__ float sbuf[WAVES][MTILE][NHEAD][CONDN];        // scores -> attnw
  __shared__ float gpart[WAVES][GROUPS][2];

  const int lane = threadIdx.x & 31;
  const int wave = threadIdx.x >> 5;
  const int b     = blockIdx.x / BLOCKS_PER_B;
  const int vbase = (blockIdx.x % BLOCKS_PER_B) * VOX_PER_BLOCK + wave * MTILE;
  const int m  = lane & 15;
  const int hi = lane >> 4;

  // Warm L2/WGP$ with the weight matrices (gfx1250 global_prefetch_b8).
  {
    const int t = threadIdx.x;
#pragma unroll
    for (int i = 0; i < 4; ++i) {
      __builtin_prefetch(wq + t * 128 + i * 32, 0, 1);
      __builtin_prefetch(wk + t * 128 + i * 32, 0, 1);
      __builtin_prefetch(wv + t * 128 + i * 32, 0, 1);
      __builtin_prefetch(wo + t * 128 + i * 32, 0, 1);
    }
  }

  float* qT = lds_q[wave];
  float* oT = lds_o[wave];

  // ---- Q phase: x-frags loaded once, reused for all 8 head tiles ----
  {
    v2f ax[32];
    load_act_frags(dec + (size_t)b * CCH * VOX + vbase, m, hi, ax);
#pragma unroll 1
    for (int nt = 0; nt < NHEAD; ++nt) {
      const int co0 = nt * DHEAD;
      v2f bw[32];
      load_w_frags(wq, co0, m, hi, bw);
      v8f qf = gemm_frags(ax, bw);
      const float bias = bq[co0 + m];
#pragma unroll
      for (int r = 0; r < 8; ++r)
        qT[(r + 8 * hi) * LDSPITCH + co0 + m] = qf[r] + bias;
    }
  }

  // ---- K phase: cond-outer so each y-tile is loaded once; scores -> sbuf ----
#pragma unroll 1
  for (int j = 0; j < CONDN; ++j) {
    v2f ay[32];
    load_act_frags(skip + (size_t)(b * CONDN + j) * CCH * VOX + vbase, m, hi, ay);
#pragma unroll 1
    for (int nt = 0; nt < NHEAD; ++nt) {
      const int co0 = nt * DHEAD;
      v2f bw[32];
      load_w_frags(wk, co0, m, hi, bw);
      v8f kf = gemm_frags(ay, bw);
      const float bias = bk[co0 + m];
#pragma unroll
      for (int r = 0; r < 8; ++r) {
        const float qv = qT[(r + 8 * hi) * LDSPITCH + co0 + m];
        float sc = red_half16(qv * (kf[r] + bias));  // dot over 16 head ch
        if (m == r) sbuf[wave][r + 8 * hi][nt][j] = sc;
      }
    }
  }

  // ---- softmax over the 4 keys (wave-private, LDS in-order per wave) ----
#pragma unroll
  for (int t = 0; t < 4; ++t) {
    const int e   = lane + 32 * t;    // 16 vox * 8 heads = 128 entries
    const int vox = e >> 3;
    const int nt  = e & 7;
    float s0 = sbuf[wave][vox][nt][0];
    float s1 = sbuf[wave][vox][nt][1];
    float s2 = sbuf[wave][vox][nt][2];
    float s3 = sbuf[wave][vox][nt][3];
    const float mx = fmaxf(fmaxf(s0, s1), fmaxf(s2, s3));
    s0 = __expf(0.25f * (s0 - mx));   // scale 1/sqrt(DHEAD) = 0.25
    s1 = __expf(0.25f * (s1 - mx));
    s2 = __expf(0.25f * (s2 - mx));
    s3 = __expf(0.25f * (s3 - mx));
    const float inv = 1.f / (s0 + s1 + s2 + s3);
    sbuf[wave][vox][nt][0] = s0 * inv;
    sbuf[wave][vox][nt][1] = s1 * inv;
    sbuf[wave][vox][nt][2] = s2 * inv;
    sbuf[wave][vox][nt][3] = s3 * inv;
  }

  // ---- V phase: cond-outer again; accumulate head outputs into lds_o ----
#pragma unroll 1
  for (int j = 0; j < CONDN; ++j) {
    v2f ay[32];
    load_act_frags(skip + (size_t)(b * CONDN + j) * CCH * VOX + vbase, m, hi, ay);
#pragma unroll 1
    for (int nt = 0; nt < NHEAD; ++nt) {
      const int co0 = nt * DHEAD;
      v2f bw[32];
      load_w_frags(wv, co0, m, hi, bw);
      v8f vf = gemm_frags(ay, bw);
      const float bias = bv[co0 + m];
#pragma unroll
      for (int r = 0; r < 8; ++r) {
        const float aw  = sbuf[wave][r + 8 * hi][nt][j];  // LDS broadcast
        const float cio = aw * (vf[r] + bias);
        float* slot = &oT[(r + 8 * hi) * LDSPITCH + co0 + m];
        if (j == 0) *slot = cio;
        else        *slot += cio;      // same-wave LDS RMW, in order
      }
    }
  }
  __syncthreads();

  // ---- O phase: out-proj from LDS stash + bias + residual + GN partials ----
  {
    v2f ao[32];
    load_lds_frags(oT, m, hi, ao);
#pragma unroll 1
    for (int nt = 0; nt < NHEAD; ++nt) {   // nt == GroupNorm group index
      const int co0 = nt * DHEAD;
      v2f bw[32];
      load_w_frags(wo, co0, m, hi, bw);
      v8f of = gemm_frags(ao, bw);
      const float bias = bo[co0 + m];
      float s1 = 0.f, s2 = 0.f;
#pragma unroll
      for (int r = 0; r < 8; ++r) {
        const int vox = r + 8 * hi;
        const size_t idx = (size_t)b * CCH * VOX + (size_t)(co0 + m) * VOX
                         + (size_t)(vbase + vox);
        const float val = of[r] + bias + dec[idx];
        out[idx] = val;
        s1 += val; s2 += val * val;
      }
      s1 = red_wave32(s1);
      s2 = red_wave32(s2);
      if (lane == 0) { gpart[wave][nt][0] = s1; gpart[wave][nt][1] = s2; }
    }
  }
  __syncthreads();

  if (threadIdx.x < GROUPS) {            // fixed-order combine: deterministic
    float s1 = 0.f, s2 = 0.f;
    for (int w = 0; w < WAVES; ++w) {
      s1 += gpart[w][threadIdx.x][0];
      s2 += gpart[w][threadIdx.x][1];
    }
    partials[((size_t)blockIdx.x * GROUPS + threadIdx.x) * 2 + 0] = s1;
    partials[((size_t)blockIdx.x * GROUPS + threadIdx.x) * 2 + 1] = s2;
  }
}

// ---------------------------------------------------------------------------
// Kernel 2: deterministic tree-reduce of block partials -> (mean, rstd)
// ---------------------------------------------------------------------------
__global__ __launch_bounds__(256) void gn_stats_kernel(
    const float* __restrict__ partials, float* __restrict__ stats) {
  __shared__ float s1s[256], s2s[256];
  const int bg = blockIdx.x;                 // b*GROUPS + g
  const int b = bg / GROUPS, g = bg % GROUPS;
  const int t = threadIdx.x;
  float s1 = 0.f, s2 = 0.f;
  for (int bb = t; bb < BLOCKS_PER_B; bb += 256) {
    const size_t blk = (size_t)b * BLOCKS_PER_B + bb;
    s1 += partials[(blk * GROUPS + g) * 2 + 0];
    s2 += partials[(blk * GROUPS + g) * 2 + 1];
  }
  s1s[t] = s1; s2s[t] = s2;
  __syncthreads();
  for (int off = 128; off > 0; off >>= 1) {
    if (t < off) { s1s[t] += s1s[t + off]; s2s[t] += s2s[t + off]; }
    __syncthreads();
  }
  if (t == 0) {
    const float cnt  = (float)((size_t)GSZ * VOX);
    const float mean = s1s[0] / cnt;
    const float var  = s2s[0] / cnt - mean * mean;
    stats[bg * 2 + 0] = mean;
    stats[bg * 2 + 1] = rsqrtf(var + EPS);
  }
}

// ---------------------------------------------------------------------------
// Kernel 3: normalize + affine, in place on `out`, float4 vectorized
// ---------------------------------------------------------------------------
__global__ __launch_bounds__(256) void gn_apply_kernel(
    float* __restrict__ out, const float* __restrict__ stats,
    const float* __restrict__ gamma, const float* __restrict__ beta) {
  const size_t total4 = (size_t)BATCH * CCH * VOX / 4;
  const size_t i4 = (size_t)blockIdx.x * blockDim.x + threadIdx.x;
  if (i4 >= total4) return;
  const size_t flat = i4 * 4;
  const int c = (int)((flat / VOX) % CCH);
  const int b = (int)(flat / ((size_t)CCH * VOX));
  const int g = c / GSZ;
  const float mean = stats[(b * GROUPS + g) * 2 + 0];
  const float rstd = stats[(b * GROUPS + g) * 2 + 1];
  const float ga = gamma[c], be = beta[c];
  float4 v = *(float4*)(out + flat);
  v.x = (v.x - mean) * rstd * ga + be;
  v.y = (v.y - mean) * rstd * ga + be;
  v.z = (v.z - mean) * rstd * ga + be;
  v.w = (v.w - mean) * rstd * ga + be;
  *(float4*)(out + flat) = v;
}

extern "C" void kernel_launch(void* const* d_in, const int* in_sizes, int n_in,
                              void* d_out, int out_size, void* d_ws, size_t ws_size,
                              hipStream_t stream) {
  const float* skip  = (const float*)d_in[0];
  const float* dec   = (const float*)d_in[1];
  const float* wq    = (const float*)d_in[2];
  const float* wk    = (const float*)d_in[3];
  const float* wv    = (const float*)d_in[4];
  const float* bq    = (const float*)d_in[5];
  const float* bk    = (const float*)d_in[6];
  const float* bv    = (const float*)d_in[7];
  const float* wo    = (const float*)d_in[8];
  const float* bo    = (const float*)d_in[9];
  const float* gamma = (const float*)d_in[10];
  const float* beta  = (const float*)d_in[11];
  float* out = (float*)d_out;

  float* partials = (float*)d_ws;                             // 1024*8*2 floats
  float* stats    = partials + (size_t)NBLOCKS * GROUPS * 2;  // 32 floats

  fused_attn_kernel<<<NBLOCKS, 128, 0, stream>>>(
      skip, dec, wq, wk, wv, bq, bk, bv, wo, bo, out, partials);
  gn_stats_kernel<<<BATCH * GROUPS, 256, 0, stream>>>(partials, stats);
  const size_t total4 = (size_t)BATCH * CCH * VOX / 4;
  gn_apply_kernel<<<(int)((total4 + 255) / 256), 256, 0, stream>>>(
      out, stats, gamma, beta);
}